// GCN_84920093376750
// MI455X (gfx1250) — compile-verified
//
#include <hip/hip_runtime.h>

#define NFEAT 13
#define HDIM  128

typedef float v2f __attribute__((ext_vector_type(2)));
typedef float v8f __attribute__((ext_vector_type(8)));

// ---------------- degree / norm ----------------
__global__ void k_deg_init(float* deg, int n) {
  int i = blockIdx.x * blockDim.x + threadIdx.x;
  if (i < n) deg[i] = 1.0f;  // self loop
}

__global__ void k_deg_edges(const int* __restrict__ dst, float* deg, int ne) {
  int e = blockIdx.x * blockDim.x + threadIdx.x;
  if (e < ne) atomicAdd(&deg[dst[e]], 1.0f);
}

__global__ void k_deg_to_dinv(float* deg, int n) {
  int i = blockIdx.x * blockDim.x + threadIdx.x;
  if (i < n) deg[i] = rsqrtf(deg[i]);   // deg >= 1 always
}

// ---------------- layer-0 GEMM (K=13, plain VALU) ----------------
// hs[i][j] = (sum_k x[i][k] * W0[k][j]) * dinv[i]; written to out AND outC (self-loop seed)
__global__ void k_gemm13(const float* __restrict__ x, const float* __restrict__ W0,
                         const float* __restrict__ dinv, float* __restrict__ out,
                         float* __restrict__ outC, int n) {
  __shared__ float sW[NFEAT * HDIM];
  int tid = threadIdx.x;                      // blockDim = 128
  for (int i = tid; i < NFEAT * HDIM; i += HDIM) sW[i] = W0[i];
  __syncthreads();
  long i0 = (long)blockIdx.x * 8;
  for (int ii = 0; ii < 8; ii++) {
    long i = i0 + ii;
    if (i >= n) return;
    float acc = 0.0f;
#pragma unroll
    for (int k = 0; k < NFEAT; k++) acc += x[i * NFEAT + k] * sW[k * HDIM + tid];
    float v = acc * dinv[i];
    out[i * HDIM + tid]  = v;
    outC[i * HDIM + tid] = v;
  }
}

// ---------------- 128x128 GEMM via V_WMMA_F32_16X16X4_F32 ----------------
// hs[i][:] = (A[i][:] @ W[128x128]) * dinv[i]; written to out AND outC (self-loop seed)
// W staged in LDS K-pair-interleaved: sWp[k/2][n] = (W[k][n], W[k+1][n]) so each B
// operand is one naturally aligned ds_load_b64 straight into an even VGPR pair.
// Inner loop: batch-load all 8 B operands (deep DScnt pipeline), then 8 WMMAs.
__global__ void k_gemm128_wmma(const float* __restrict__ A, const float* __restrict__ W,
                               const float* __restrict__ dinv, float* __restrict__ out,
                               float* __restrict__ outC, int ntiles) {
  __shared__ v2f sWp[(HDIM / 2) * HDIM];      // 64 KB of the 320 KB WGP LDS
  int tid = threadIdx.x;                      // blockDim = 128 (4 waves)
  for (int idx = tid; idx < (HDIM / 2) * HDIM; idx += HDIM) {
    int kk = idx >> 7;                        // K pair index 0..63
    int n  = idx & 127;
    v2f p;
    p.x = W[(2 * kk + 0) * HDIM + n];
    p.y = W[(2 * kk + 1) * HDIM + n];
    sWp[idx] = p;
  }
  __syncthreads();

  int wave = tid >> 5;
  int lane = tid & 31;
  int lo = lane & 15;        // M index (A) / N index (B, C)
  int hi = lane >> 4;        // selects K pair per ISA VGPR layout

  int tileRow = blockIdx.x * 4 + wave;        // wave-uniform guard: EXEC stays all-1s
  if (tileRow >= ntiles) return;
  int row0 = tileRow * 16;

  v8f acc[8] = {};
  const float* Arow = A + (long)(row0 + lo) * HDIM;

  for (int k0 = 0; k0 < HDIM; k0 += 4) {
    // A 16x4 f32: lanes0-15 -> K=k0,k0+1 ; lanes16-31 -> K=k0+2,k0+3
    v2f a = *(const v2f*)(Arow + k0 + 2 * hi);
    const v2f* brow = sWp + ((k0 >> 1) + hi) * HDIM;  // pair row k0/2 (+1 for hi half)
    v2f b[8];
#pragma unroll
    for (int t = 0; t < 8; t++) b[t] = brow[t * 16 + lo];   // issue all 8 ds loads
#pragma unroll
    for (int t = 0; t < 8; t++) {
      acc[t] = __builtin_amdgcn_wmma_f32_16x16x4_f32(
          /*neg_a=*/false, a, /*neg_b=*/false, b[t],
          /*c_mod=*/(short)0, acc[t], /*reuse_a=*/false, /*reuse_b=*/false);
    }
  }

  // C/D layout: VGPR v holds M = v + 8*hi, column = n0 + lo
#pragma unroll
  for (int t = 0; t < 8; t++) {
    int n0 = t * 16;
#pragma unroll
    for (int v = 0; v < 8; v++) {
      int r = row0 + v + 8 * hi;
      float val = acc[t][v] * dinv[r];
      out[(long)r * HDIM + n0 + lo]  = val;
      outC[(long)r * HDIM + n0 + lo] = val;
    }
  }
}

// ---------------- aggregation ----------------
// per edge: 32 lanes x float4 gather of hs[src], atomic add into agg[dst] (hits 192MB L2)
__global__ void k_edge_agg(const int* __restrict__ src, const int* __restrict__ dst,
                           const float* __restrict__ B, float* __restrict__ C, int ne) {
  long t = (long)blockIdx.x * blockDim.x + threadIdx.x;
  long e = t >> 5;
  int lane = (int)(t & 31);
  if (e >= ne) return;
  int s = src[e], d = dst[e];
  float4 v = *(const float4*)(B + (long)s * HDIM + lane * 4);
  float* cp = C + (long)d * HDIM + lane * 4;
  atomicAdd(cp + 0, v.x);
  atomicAdd(cp + 1, v.y);
  atomicAdd(cp + 2, v.z);
  atomicAdd(cp + 3, v.w);
}

__global__ void k_epilogue(const float* __restrict__ C, const float* __restrict__ dinv,
                           const float* __restrict__ bias, float* __restrict__ A,
                           long total, int relu) {
  long t = (long)blockIdx.x * blockDim.x + threadIdx.x;
  if (t >= total) return;
  long i = t >> 7;
  int j = (int)(t & 127);
  float v = C[t] * dinv[i] + bias[j];
  if (relu) v = fmaxf(v, 0.0f);
  A[t] = v;
}

// ---------------- pooling ----------------
__global__ void k_zero(float* p, long n) {
  long i = (long)blockIdx.x * blockDim.x + threadIdx.x;
  if (i < n) p[i] = 0.0f;
}

__global__ void k_pool_accum(const float* __restrict__ H, const int* __restrict__ batch,
                             float* pooled, float* cnt, long total) {
  long t = (long)blockIdx.x * blockDim.x + threadIdx.x;
  if (t >= total) return;
  long i = t >> 7;
  int j = (int)(t & 127);
  int g = batch[i];
  atomicAdd(&pooled[(long)g * HDIM + j], H[t]);
  if (j == 0) atomicAdd(&cnt[g], 1.0f);
}

__global__ void k_pool_div(float* pooled, const float* cnt, long total) {
  long t = (long)blockIdx.x * blockDim.x + threadIdx.x;
  if (t >= total) return;
  pooled[t] /= fmaxf(cnt[t >> 7], 1.0f);
}

// ---------------- MLP head ----------------
// one block per graph; blockDim = M; in-row cached in LDS
__global__ void k_fc(const float* __restrict__ in, const float* __restrict__ W,
                     const float* __restrict__ bias, float* __restrict__ out,
                     int K, int M, int relu) {
  __shared__ float srow[256];
  int g = blockIdx.x;
  int tid = threadIdx.x;
  for (int k = tid; k < K; k += M) srow[k] = in[(long)g * K + k];
  __syncthreads();
  float acc = bias[tid];
  for (int k = 0; k < K; k++) acc += srow[k] * W[k * M + tid];
  if (relu) acc = fmaxf(acc, 0.0f);
  out[(long)g * M + tid] = acc;
}

__global__ void k_fc_out(const float* __restrict__ in, const float* __restrict__ Wo,
                         const float* __restrict__ bo, float* __restrict__ out) {
  __shared__ float red[HDIM];
  int g = blockIdx.x;
  int tid = threadIdx.x;  // 128
  red[tid] = in[(long)g * HDIM + tid] * Wo[tid];
  __syncthreads();
  for (int s = 64; s > 0; s >>= 1) {
    if (tid < s) red[tid] += red[tid + s];
    __syncthreads();
  }
  if (tid == 0) out[g] = red[0] + bo[0];
}

// ---------------- driver ----------------
extern "C" void kernel_launch(void* const* d_in, const int* in_sizes, int n_in,
                              void* d_out, int out_size, void* d_ws, size_t ws_size,
                              hipStream_t stream) {
  const float* x     = (const float*)d_in[0];
  const int*   ei    = (const int*)d_in[1];
  const int*   batch = (const int*)d_in[2];
  const float* W0 = (const float*)d_in[3];   const float* b0 = (const float*)d_in[4];
  const float* W1 = (const float*)d_in[5];   const float* b1 = (const float*)d_in[6];
  const float* W2 = (const float*)d_in[7];   const float* b2 = (const float*)d_in[8];
  const float* Wf0 = (const float*)d_in[9];  const float* bf0 = (const float*)d_in[10];
  const float* Wf1 = (const float*)d_in[11]; const float* bf1 = (const float*)d_in[12];
  const float* Wo  = (const float*)d_in[13]; const float* bo  = (const float*)d_in[14];
  float* out = (float*)d_out;

  const int N = in_sizes[0] / NFEAT;     // 100000
  const int E = in_sizes[1] / 2;         // 1600000
  const int G = out_size;                // 2000
  const int* esrc = ei;
  const int* edst = ei + E;

  // workspace carve-up (256B aligned)
  char* w = (char*)d_ws;
  auto alloc = [&](size_t bytes) -> float* {
    float* p = (float*)w;
    w += (bytes + 255) & ~(size_t)255;
    return p;
  };
  float* dinv   = alloc((size_t)N * 4);
  float* bufA   = alloc((size_t)N * HDIM * 4);  // layer input activations
  float* bufB   = alloc((size_t)N * HDIM * 4);  // hs = (A@W)*dinv
  float* bufC   = alloc((size_t)N * HDIM * 4);  // aggregate (seeded with hs = self loop)
  float* pooled = alloc((size_t)G * HDIM * 4);
  float* fcb0   = alloc((size_t)G * 256 * 4);
  float* fcb1   = alloc((size_t)G * HDIM * 4);
  float* cnt    = alloc((size_t)G * 4);
  (void)ws_size; (void)n_in;

  const long total = (long)N * HDIM;
  const int  ntiles = N / 16;                       // 6250 (N divisible by 16)
  dim3 blk256(256), blk128(128);
  int gN    = (N + 255) / 256;
  int gE    = (E + 255) / 256;
  int gTot  = (int)((total + 255) / 256);
  int gEdge = (int)(((long)E * 32 + 255) / 256);
  int gGemm = (ntiles + 3) / 4;

  // symmetric norm: dinv = rsqrt(in-degree + 1)
  k_deg_init<<<gN, blk256, 0, stream>>>(dinv, N);
  k_deg_edges<<<gE, blk256, 0, stream>>>(edst, dinv, E);
  k_deg_to_dinv<<<gN, blk256, 0, stream>>>(dinv, N);

  // ---- layer 0 ----
  k_gemm13<<<(N + 7) / 8, blk128, 0, stream>>>(x, W0, dinv, bufB, bufC, N);
  k_edge_agg<<<gEdge, blk256, 0, stream>>>(esrc, edst, bufB, bufC, E);
  k_epilogue<<<gTot, blk256, 0, stream>>>(bufC, dinv, b0, bufA, total, 1);

  // ---- layer 1 (WMMA) ----
  k_gemm128_wmma<<<gGemm, blk128, 0, stream>>>(bufA, W1, dinv, bufB, bufC, ntiles);
  k_edge_agg<<<gEdge, blk256, 0, stream>>>(esrc, edst, bufB, bufC, E);
  k_epilogue<<<gTot, blk256, 0, stream>>>(bufC, dinv, b1, bufA, total, 1);

  // ---- layer 2 (WMMA, no relu) ----
  k_gemm128_wmma<<<gGemm, blk128, 0, stream>>>(bufA, W2, dinv, bufB, bufC, ntiles);
  k_edge_agg<<<gEdge, blk256, 0, stream>>>(esrc, edst, bufB, bufC, E);
  k_epilogue<<<gTot, blk256, 0, stream>>>(bufC, dinv, b2, bufA, total, 0);

  // ---- global mean pool ----
  long pTot = (long)G * HDIM;
  k_zero<<<(int)((pTot + 255) / 256), blk256, 0, stream>>>(pooled, pTot);
  k_zero<<<(G + 255) / 256, blk256, 0, stream>>>(cnt, G);
  k_pool_accum<<<gTot, blk256, 0, stream>>>(bufA, batch, pooled, cnt, total);
  k_pool_div<<<(int)((pTot + 255) / 256), blk256, 0, stream>>>(pooled, cnt, pTot);

  // ---- MLP head ----
  k_fc<<<G, 256, 0, stream>>>(pooled, Wf0, bf0, fcb0, 128, 256, 1);
  k_fc<<<G, 128, 0, stream>>>(fcb0, Wf1, bf1, fcb1, 256, 128, 1);
  k_fc_out<<<G, 128, 0, stream>>>(fcb1, Wo, bo, out);
}